// SelfAttention_21483426414660
// MI455X (gfx1250) — compile-verified
//
#include <hip/hip_runtime.h>
#include <cmath>
#include <stdint.h>

// Self-attention (fp32), MI455X / gfx1250, wave32, WMMA f32 16x16x4.
// Stage 1: qkv = x @ W_qkv + b_qkv   -> Q(scaled),K,V in ws, head-major [b,h,s,d]
// Stage 2: flash attention per (b,h) -> attn ws [b,h,s,d]
// Stage 3: out = attn @ W_out + b_out

#define BATCH 4
#define SEQ   2048
#define DM    1024
#define NH    16
#define HD    64
#define QK_SCALE 0.125f   // 1/sqrt(64)

typedef __attribute__((ext_vector_type(2))) float v2f;
typedef __attribute__((ext_vector_type(8))) float v8f;

__device__ __forceinline__ v8f wmma_f32(v2f a, v2f b, v8f c) {
  // (neg_a, A, neg_b, B, c_mod, C, reuse_a, reuse_b)
  return __builtin_amdgcn_wmma_f32_16x16x4_f32(false, a, false, b, (short)0, c,
                                               false, false);
}

// ---- CDNA5 async global->LDS staging (ASYNCcnt-tracked), with fallback -----
#if __has_builtin(__builtin_amdgcn_global_load_async_to_lds_b128)
#define HAVE_ASYNC_LDS 1
#else
#define HAVE_ASYNC_LDS 0
#endif

typedef int b128_t __attribute__((vector_size(16)));   // v4i payload type

__device__ __forceinline__ void stage_b128(const float* g, float* l) {
#if HAVE_ASYNC_LDS
  // flat->as(1) is numerically identity; flat shared low 32 bits == LDS offset
  __builtin_amdgcn_global_load_async_to_lds_b128(
      (__attribute__((address_space(1))) b128_t*)(uintptr_t)g,
      (__attribute__((address_space(3))) b128_t*)(uint32_t)(uintptr_t)l, 0, 0);
#else
  *(float4*)l = *(const float4*)g;
#endif
}

__device__ __forceinline__ void stage_wait() {
#if HAVE_ASYNC_LDS
#if __has_builtin(__builtin_amdgcn_s_wait_asynccnt)
  __builtin_amdgcn_s_wait_asynccnt(0);
#else
  asm volatile("s_wait_asynccnt 0x0" ::: "memory");
#endif
#endif
}

// ---------------------------------------------------------------------------
// Kernel 1: QKV projection. grid=(24,128), block=256 (8 waves).
// Wave w computes FOUR stacked 16x16 tiles (64 rows x 16 cols): each B-pair
// feeds 4 WMMAs -> 4x fewer global B loads. A chunk (64 x 128) in LDS,
// padded stride 132 -> conflict-free ((row*4+kb)%64 distinct across lanes).
// ---------------------------------------------------------------------------
__global__ __launch_bounds__(256) void qkv_gemm_kernel(
    const float* __restrict__ x, const float* __restrict__ W,
    const float* __restrict__ bias, float* __restrict__ qws,
    float* __restrict__ kws, float* __restrict__ vws) {
  __shared__ float As[64 * 132];
  const int wave = threadIdx.x >> 5;
  const int lane = threadIdx.x & 31;
  const int half = lane >> 4;
  const int ln   = lane & 15;
  const int row0 = blockIdx.y * 64;
  const int col  = (blockIdx.x * 8 + wave) * 16 + ln;

  v8f c0 = {}, c1 = {}, c2 = {}, c3 = {};
  for (int kc = 0; kc < 8; ++kc) {
    // stage x[row0..+63][kc*128..+127] : 2048 float4, 8 per thread
#pragma unroll
    for (int it = 0; it < 8; ++it) {
      int idx = threadIdx.x + it * 256;   // float4 index 0..2047
      int r   = idx >> 5;                 // 32 float4 per row
      int c4  = idx & 31;
      stage_b128(x + (size_t)(row0 + r) * DM + kc * 128 + c4 * 4,
                 &As[r * 132 + c4 * 4]);
    }
    stage_wait();
    __syncthreads();
#pragma unroll
    for (int j = 0; j < 32; ++j) {
      int kb = 4 * j + 2 * half;
      int kg = kc * 128 + kb;
      v2f b;
      b.x = W[(size_t)kg * (3 * DM) + col];
      b.y = W[(size_t)(kg + 1) * (3 * DM) + col];
      v2f a;
      a.x = As[(0 * 16 + ln) * 132 + kb]; a.y = As[(0 * 16 + ln) * 132 + kb + 1];
      c0 = wmma_f32(a, b, c0);
      a.x = As[(1 * 16 + ln) * 132 + kb]; a.y = As[(1 * 16 + ln) * 132 + kb + 1];
      c1 = wmma_f32(a, b, c1);
      a.x = As[(2 * 16 + ln) * 132 + kb]; a.y = As[(2 * 16 + ln) * 132 + kb + 1];
      c2 = wmma_f32(a, b, c2);
      a.x = As[(3 * 16 + ln) * 132 + kb]; a.y = As[(3 * 16 + ln) * 132 + kb + 1];
      c3 = wmma_f32(a, b, c3);
    }
    __syncthreads();
  }

  // scatter with bias into head-major Q/K/V; Q pre-scaled by 1/sqrt(hd)
  const float bb  = bias[col];
  const int which = col >> 10;   // 0=Q, 1=K, 2=V
  const int cc    = col & (DM - 1);
  const int h     = cc >> 6;
  const int d     = cc & (HD - 1);
  float* dst = (which == 0) ? qws : ((which == 1) ? kws : vws);
  const float mul = (which == 0) ? QK_SCALE : 1.0f;
  const v8f acc[4] = {c0, c1, c2, c3};
#pragma unroll
  for (int mt = 0; mt < 4; ++mt) {
#pragma unroll
    for (int i = 0; i < 8; ++i) {
      int r    = row0 + mt * 16 + i + 8 * half;
      int bidx = r / SEQ;
      int s    = r & (SEQ - 1);
      dst[(((size_t)bidx * NH + h) * SEQ + s) * HD + d] = (acc[mt][i] + bb) * mul;
    }
  }
}

// ---------------------------------------------------------------------------
// Kernel 2: flash attention. grid=(SEQ/128, BATCH*NH), block=256 (8 waves).
// Wave w owns q-tile (blockIdx.x*8+w): 16 q rows x 64 head dims.
// K/V tiles (16x64) async-staged to LDS (stride 68); P bounced through LDS
// for C-layout -> A-layout. Online softmax via 16-lane shfl_xor (C layout
// keeps each score row within one lane-half, so reductions are cheap).
// ---------------------------------------------------------------------------
__global__ __launch_bounds__(256) void attn_kernel(
    const float* __restrict__ qws, const float* __restrict__ kws,
    const float* __restrict__ vws, float* __restrict__ ows) {
  __shared__ float Ks[16 * 68];
  __shared__ float Vs[16 * 68];
  __shared__ float Ps[8][16 * 18];
  const int wave  = threadIdx.x >> 5;
  const int lane  = threadIdx.x & 31;
  const int half  = lane >> 4;
  const int ln    = lane & 15;
  const int bh    = blockIdx.y;
  const int qtile = blockIdx.x * 8 + wave;
  const size_t base = (size_t)bh * SEQ * HD;

  // Q rows in A-layout registers (already scaled by 1/sqrt(hd))
  v2f qa[16];
  {
    const float* qrow = qws + base + (size_t)(qtile * 16 + ln) * HD;
#pragma unroll
    for (int j = 0; j < 16; ++j)
      qa[j] = *(const v2f*)(qrow + 4 * j + 2 * half);
  }

  v8f acc0 = {}, acc1 = {}, acc2 = {}, acc3 = {};
  float mrow[8], lrow[8];
#pragma unroll
  for (int i = 0; i < 8; ++i) { mrow[i] = -__builtin_inff(); lrow[i] = 0.f; }

  const int last = blockIdx.x * 8 + 7;
  const int lr  = threadIdx.x >> 4;   // staging row 0..15
  const int lc4 = threadIdx.x & 15;   // staging float4 col 0..15

  for (int kt = 0; kt <= last; ++kt) {
    {
      const size_t off = base + (size_t)(kt * 16 + lr) * HD + lc4 * 4;
      stage_b128(kws + off, &Ks[lr * 68 + lc4 * 4]);
      stage_b128(vws + off, &Vs[lr * 68 + lc4 * 4]);
    }
    stage_wait();
    __syncthreads();
    if (kt <= qtile) {
      // scores S = Q @ K^T : 16 WMMAs over head dim 64
      v8f s8 = {};
#pragma unroll
      for (int j = 0; j < 16; ++j) {
        int kb = 4 * j + 2 * half;
        v2f b;
        b.x = Ks[ln * 68 + kb];       // B col = key index (ln)
        b.y = Ks[ln * 68 + kb + 1];
        s8 = wmma_f32(qa[j], b, s8);
      }
      // causal mask on diagonal tile: key(ln) > qrow(m) -> -inf
      if (kt == qtile) {
#pragma unroll
        for (int i = 0; i < 8; ++i) {
          int m = i + 8 * half;
          s8[i] = (ln > m) ? -__builtin_inff() : s8[i];
        }
      }
      // online softmax: row reductions across 16 lanes of each half
#pragma unroll
      for (int i = 0; i < 8; ++i) {
        float v = s8[i];
        v = fmaxf(v, __shfl_xor(v, 1, 32));
        v = fmaxf(v, __shfl_xor(v, 2, 32));
        v = fmaxf(v, __shfl_xor(v, 4, 32));
        v = fmaxf(v, __shfl_xor(v, 8, 32));
        float nm    = fmaxf(mrow[i], v);
        float alpha = __expf(mrow[i] - nm);
        mrow[i] = nm;
        lrow[i] *= alpha;
        acc0[i] *= alpha; acc1[i] *= alpha; acc2[i] *= alpha; acc3[i] *= alpha;
        float pe = __expf(s8[i] - nm);
        float ps = pe;
        ps += __shfl_xor(ps, 1, 32);
        ps += __shfl_xor(ps, 2, 32);
        ps += __shfl_xor(ps, 4, 32);
        ps += __shfl_xor(ps, 8, 32);
        lrow[i] += ps;
        Ps[wave][(i + 8 * half) * 18 + ln] = pe;   // C-layout -> LDS
      }
      // O += P @ V : P in A-layout from LDS, V as B operand
#pragma unroll
      for (int jj = 0; jj < 4; ++jj) {
        int kb = 4 * jj + 2 * half;
        v2f a;
        a.x = Ps[wave][ln * 18 + kb];
        a.y = Ps[wave][ln * 18 + kb + 1];
        v2f b0, b1, b2, b3;
        b0.x = Vs[kb * 68 + ln];        b0.y = Vs[(kb + 1) * 68 + ln];
        b1.x = Vs[kb * 68 + 16 + ln];   b1.y = Vs[(kb + 1) * 68 + 16 + ln];
        b2.x = Vs[kb * 68 + 32 + ln];   b2.y = Vs[(kb + 1) * 68 + 32 + ln];
        b3.x = Vs[kb * 68 + 48 + ln];   b3.y = Vs[(kb + 1) * 68 + 48 + ln];
        acc0 = wmma_f32(a, b0, acc0);
        acc1 = wmma_f32(a, b1, acc1);
        acc2 = wmma_f32(a, b2, acc2);
        acc3 = wmma_f32(a, b3, acc3);
      }
    }
    __syncthreads();
  }

  // normalize and store head-major output
#pragma unroll
  for (int i = 0; i < 8; ++i) {
    int m = i + 8 * half;
    float inv = 1.0f / lrow[i];
    size_t o = base + (size_t)(qtile * 16 + m) * HD;
    ows[o + ln]      = acc0[i] * inv;
    ows[o + 16 + ln] = acc1[i] * inv;
    ows[o + 32 + ln] = acc2[i] * inv;
    ows[o + 48 + ln] = acc3[i] * inv;
  }
}

// ---------------------------------------------------------------------------
// Kernel 3: output projection. grid=(8,128), block=256. Same 4x M-blocked
// skeleton; A rows gathered from head-major attn ws during LDS staging.
// ---------------------------------------------------------------------------
__global__ __launch_bounds__(256) void out_gemm_kernel(
    const float* __restrict__ attn, const float* __restrict__ W,
    const float* __restrict__ bias, float* __restrict__ out) {
  __shared__ float As[64 * 132];
  const int wave = threadIdx.x >> 5;
  const int lane = threadIdx.x & 31;
  const int half = lane >> 4;
  const int ln   = lane & 15;
  const int row0 = blockIdx.y * 64;
  const int col  = (blockIdx.x * 8 + wave) * 16 + ln;

  v8f c0 = {}, c1 = {}, c2 = {}, c3 = {};
  for (int kc = 0; kc < 8; ++kc) {
#pragma unroll
    for (int it = 0; it < 8; ++it) {
      int idx  = threadIdx.x + it * 256;
      int r    = idx >> 5;
      int c4   = idx & 31;
      int ccol = c4 * 4;               // within-chunk col (multiple of 4)
      int k    = kc * 128 + ccol;      // global k in [0,1024)
      int h    = k >> 6;
      int d    = k & (HD - 1);
      int grow = row0 + r;
      int bidx = grow / SEQ;
      int s    = grow & (SEQ - 1);
      stage_b128(attn + (((size_t)bidx * NH + h) * SEQ + s) * HD + d,
                 &As[r * 132 + ccol]);
    }
    stage_wait();
    __syncthreads();
#pragma unroll
    for (int j = 0; j < 32; ++j) {
      int kb = 4 * j + 2 * half;
      int kg = kc * 128 + kb;
      v2f b;
      b.x = W[(size_t)kg * DM + col];
      b.y = W[(size_t)(kg + 1) * DM + col];
      v2f a;
      a.x = As[(0 * 16 + ln) * 132 + kb]; a.y = As[(0 * 16 + ln) * 132 + kb + 1];
      c0 = wmma_f32(a, b, c0);
      a.x = As[(1 * 16 + ln) * 132 + kb]; a.y = As[(1 * 16 + ln) * 132 + kb + 1];
      c1 = wmma_f32(a, b, c1);
      a.x = As[(2 * 16 + ln) * 132 + kb]; a.y = As[(2 * 16 + ln) * 132 + kb + 1];
      c2 = wmma_f32(a, b, c2);
      a.x = As[(3 * 16 + ln) * 132 + kb]; a.y = As[(3 * 16 + ln) * 132 + kb + 1];
      c3 = wmma_f32(a, b, c3);
    }
    __syncthreads();
  }
  const float bb = bias[col];
  const v8f acc[4] = {c0, c1, c2, c3};
#pragma unroll
  for (int mt = 0; mt < 4; ++mt) {
#pragma unroll
    for (int i = 0; i < 8; ++i) {
      int m = mt * 16 + i + 8 * half;
      out[(size_t)(row0 + m) * DM + col] = acc[mt][i] + bb;
    }
  }
}

// ---------------------------------------------------------------------------
extern "C" void kernel_launch(void* const* d_in, const int* in_sizes, int n_in,
                              void* d_out, int out_size, void* d_ws,
                              size_t ws_size, hipStream_t stream) {
  (void)in_sizes; (void)n_in; (void)out_size; (void)ws_size;
  const float* x     = (const float*)d_in[0];
  // d_in[1] is the boolean causal mask; causality is handled analytically.
  const float* W_qkv = (const float*)d_in[2];
  const float* b_qkv = (const float*)d_in[3];
  const float* W_out = (const float*)d_in[4];
  const float* b_out = (const float*)d_in[5];
  float* out = (float*)d_out;

  const size_t QS = (size_t)BATCH * NH * SEQ * HD;  // 8,388,608 floats
  float* q  = (float*)d_ws;
  float* k  = q + QS;
  float* v  = k + QS;
  float* at = v + QS;

  qkv_gemm_kernel<<<dim3(3 * DM / 128, BATCH * SEQ / 64), 256, 0, stream>>>(
      x, W_qkv, b_qkv, q, k, v);
  attn_kernel<<<dim3(SEQ / 128, BATCH * NH), 256, 0, stream>>>(q, k, v, at);
  out_gemm_kernel<<<dim3(DM / 128, BATCH * SEQ / 64), 256, 0, stream>>>(
      at, W_out, b_out, out);
}